// Net_47734266527818
// MI455X (gfx1250) — compile-verified
//
#include <hip/hip_runtime.h>
#include <hip/hip_bf16.h>

// ---------------------------------------------------------------------------
// 2-layer GAT forward for gfx1250 (MI455X), wave32 + V_WMMA_F32_16X16X4_F32.
// N=50000 nodes, F_in=64, layer1: 8 heads x 64ch (=512 concat), layer2: 17.
// Edge list [2,E] int32 + N self loops appended.
// Roofline: edge aggregate (~1.1GB L2-resident gather + 281M f32 atomics)
// dominates; GEMMs stay on the full-precision f32 WMMA pipe.
// ---------------------------------------------------------------------------

typedef float v2f __attribute__((ext_vector_type(2)));
typedef float v8f __attribute__((ext_vector_type(8)));

#define NEG_SLOPE 0.2f

__device__ __forceinline__ float leakyf(float v) { return v > 0.f ? v : NEG_SLOPE * v; }
__device__ __forceinline__ float eluf(float v)   { return v > 0.f ? v : (__expf(v) - 1.f); }

// IEEE-order-preserving float atomic max (buffer init to -inf).
__device__ __forceinline__ void atomicMaxF(float* addr, float v) {
    if (v >= 0.f) atomicMax((int*)addr, __float_as_int(v));
    else          atomicMin((unsigned int*)addr, __float_as_uint(v));
}

// src/dst for edge e in [0, E+N): first E from edge_index, then self loops.
__device__ __forceinline__ void edge_sd(const int* __restrict__ ei, int E, int e,
                                        int& s, int& d) {
    if (e < E) { s = ei[e]; d = ei[E + e]; }
    else       { s = e - E; d = s; }
}

// ------------------------------- init kernels ------------------------------

__global__ void k_fill(float* __restrict__ p, float v, int n) {
    int i = blockIdx.x * blockDim.x + threadIdx.x;
    if (i < n) p[i] = v;
}

// p[n*C + c] = b[c]   (bias broadcast init of the scatter accumulators)
__global__ void k_bias_init(float* __restrict__ p, const float* __restrict__ b,
                            int total, int C) {
    int i = blockIdx.x * blockDim.x + threadIdx.x;
    if (i < total) p[i] = b[i % C];
}

// Zero-pad W2 [512,17] -> W2p [512,32] so GEMM2 tiles cleanly for 16-wide WMMA.
__global__ void k_pad_w2(const float* __restrict__ W2, float* __restrict__ W2p) {
    int i = blockIdx.x * blockDim.x + threadIdx.x;
    if (i < 512 * 32) {
        int k = i >> 5, n = i & 31;
        W2p[i] = (n < 17) ? W2[k * 17 + n] : 0.f;
    }
}

// ------------------------- GEMM1: h1 = x @ W1 (f32 WMMA) -------------------
// One wave computes a 16x64 strip (1 M-tile x 4 N-tiles): each A fragment
// feeds 4 WMMAs. K=64 in steps of 4 (V_WMMA_F32_16X16X4_F32).
// A frag (16x4 f32): lanes 0-15 K={0,1}, lanes 16-31 K={2,3}, M=lane&15.
// B frag (4x16 f32): same K split, N=lane&15.
// C/D (16x16 f32, 8 VGPRs): vgpr r -> row r (+8 for upper half-wave).
__global__ void gemm1_wmma(const float* __restrict__ X,   // [N,64]
                           const float* __restrict__ W,   // [64,512]
                           float* __restrict__ H,         // [N,512]
                           int N, int Mt) {
    const int lane = threadIdx.x & 31;
    const int wave = blockIdx.x * (blockDim.x >> 5) + (threadIdx.x >> 5);
    const int mt = wave >> 3;                 // 8 groups of 4 column tiles
    const int grp = wave & 7;
    if (mt >= Mt) return;                     // wave-uniform: EXEC stays all-ones

    const int mrow = mt * 16 + (lane & 15);
    const int m    = mrow < N ? mrow : N - 1;
    const int nbase = grp * 64 + (lane & 15); // first of 4 col tiles, 16 apart
    const int kb   = (lane >> 4) * 2;         // K sub-offset per half-wave

    v8f acc0 = {}, acc1 = {}, acc2 = {}, acc3 = {};
#pragma unroll
    for (int k0 = 0; k0 < 64; k0 += 4) {
        __builtin_prefetch(&X[m * 64 + ((k0 + 8) & 63)], 0, 1);
        v2f a;
        a.x = X[m * 64 + k0 + kb];
        a.y = X[m * 64 + k0 + kb + 1];
        const float* w0 = W + (k0 + kb) * 512 + nbase;
        const float* w1 = W + (k0 + kb + 1) * 512 + nbase;
        v2f b0, b1, b2, b3;
        b0.x = w0[0];  b0.y = w1[0];
        b1.x = w0[16]; b1.y = w1[16];
        b2.x = w0[32]; b2.y = w1[32];
        b3.x = w0[48]; b3.y = w1[48];
        acc0 = __builtin_amdgcn_wmma_f32_16x16x4_f32(false, a, false, b0, (short)0, acc0, false, false);
        acc1 = __builtin_amdgcn_wmma_f32_16x16x4_f32(false, a, false, b1, (short)0, acc1, false, false);
        acc2 = __builtin_amdgcn_wmma_f32_16x16x4_f32(false, a, false, b2, (short)0, acc2, false, false);
        acc3 = __builtin_amdgcn_wmma_f32_16x16x4_f32(false, a, false, b3, (short)0, acc3, false, false);
    }

    const int row0 = mt * 16 + 8 * (lane >> 4);
#pragma unroll
    for (int r = 0; r < 8; ++r) {
        int row = row0 + r;
        if (row < N) {
            float* o = H + (size_t)row * 512 + nbase;
            o[0]  = acc0[r];
            o[16] = acc1[r];
            o[32] = acc2[r];
            o[48] = acc3[r];
        }
    }
}

// ----------------- attention coefficients layer 1: [N,8] -------------------
__global__ void alpha1_kernel(const float* __restrict__ h1,
                              const float* __restrict__ a_src,  // [8,64]
                              const float* __restrict__ a_dst,  // [8,64]
                              float* __restrict__ asrc, float* __restrict__ adst,
                              int N) {
    int t = blockIdx.x * blockDim.x + threadIdx.x;
    if (t >= N * 8) return;
    int n = t >> 3, h = t & 7;
    const float4* row = (const float4*)(h1 + (size_t)n * 512 + h * 64);
    const float4* ws  = (const float4*)(a_src + h * 64);
    const float4* wd  = (const float4*)(a_dst + h * 64);
    float s = 0.f, d = 0.f;
#pragma unroll
    for (int c = 0; c < 16; ++c) {
        float4 v = row[c], a = ws[c], b = wd[c];
        s += v.x * a.x + v.y * a.y + v.z * a.z + v.w * a.w;
        d += v.x * b.x + v.y * b.y + v.z * b.z + v.w * b.w;
    }
    asrc[t] = s; adst[t] = d;
}

// ------------------- edge logits + segment max (layer 1) -------------------
__global__ void edge_logits1(const int* __restrict__ ei, int E, int ET,
                             const float* __restrict__ asrc,
                             const float* __restrict__ adst,
                             float* __restrict__ e1, float* __restrict__ m1) {
    int t = blockIdx.x * blockDim.x + threadIdx.x;
    if (t >= ET * 8) return;
    int e = t >> 3, h = t & 7;
    int s, d; edge_sd(ei, E, e, s, d);
    float v = leakyf(asrc[s * 8 + h] + adst[d * 8 + h]);
    e1[t] = v;
    atomicMaxF(&m1[d * 8 + h], v);
}

// --------------------- exp + segment sum (layer 1) -------------------------
__global__ void edge_p1(const int* __restrict__ ei, int E, int ET,
                        float* __restrict__ e1, const float* __restrict__ m1,
                        float* __restrict__ d1) {
    int t = blockIdx.x * blockDim.x + threadIdx.x;
    if (t >= ET * 8) return;
    int e = t >> 3, h = t & 7;
    int s, d; edge_sd(ei, E, e, s, d);
    float p = __expf(e1[t] - m1[d * 8 + h]);
    e1[t] = p;
    atomicAdd(&d1[d * 8 + h], p);
}

// --------- weighted scatter aggregate (layer 1): out1 += alpha*h1[src] -----
// 2 edges per 256-thread block; 128 lanes/edge, one float4 (b128 gather) each.
__global__ void aggregate1(const int* __restrict__ ei, int E, int ET,
                           const float* __restrict__ h1,
                           const float* __restrict__ e1,
                           const float* __restrict__ d1,
                           float* __restrict__ out1) {
    int e = blockIdx.x * 2 + (threadIdx.x >> 7);
    if (e >= ET) return;
    int s, d; edge_sd(ei, E, e, s, d);
    int q = threadIdx.x & 127;                  // float4 index within 512 cols
    int h = q >> 4;                             // 16 float4 per head
    float alpha = e1[e * 8 + h] / (d1[d * 8 + h] + 1e-16f);
    float4 v = ((const float4*)(h1 + (size_t)s * 512))[q];
    float* o = out1 + (size_t)d * 512 + q * 4;
    atomicAdd(o + 0, alpha * v.x);
    atomicAdd(o + 1, alpha * v.y);
    atomicAdd(o + 2, alpha * v.z);
    atomicAdd(o + 3, alpha * v.w);
}

// ------------- GEMM2: h2 = elu(out1) @ W2p  (ELU fused in A-load) ----------
// One wave per M-tile computes both padded N-tiles; A (with ELU) loaded once.
__global__ void gemm2_wmma(const float* __restrict__ OUT1,  // [N,512] pre-ELU
                           const float* __restrict__ W2p,   // [512,32] padded
                           float* __restrict__ H2,          // [N,17]
                           int N, int Mt) {
    const int lane = threadIdx.x & 31;
    const int mt = blockIdx.x * (blockDim.x >> 5) + (threadIdx.x >> 5);
    if (mt >= Mt) return;

    const int mrow = mt * 16 + (lane & 15);
    const int m    = mrow < N ? mrow : N - 1;
    const int n16  = (lane & 15);
    const int kb   = (lane >> 4) * 2;

    v8f acc0 = {}, acc1 = {};
    for (int k0 = 0; k0 < 512; k0 += 4) {
        __builtin_prefetch(&OUT1[(size_t)m * 512 + ((k0 + 16) & 511)], 0, 1);
        v2f a, b0, b1;
        a.x = eluf(OUT1[(size_t)m * 512 + k0 + kb]);
        a.y = eluf(OUT1[(size_t)m * 512 + k0 + kb + 1]);
        const float* w0 = W2p + (k0 + kb) * 32 + n16;
        const float* w1 = W2p + (k0 + kb + 1) * 32 + n16;
        b0.x = w0[0];  b0.y = w1[0];
        b1.x = w0[16]; b1.y = w1[16];
        acc0 = __builtin_amdgcn_wmma_f32_16x16x4_f32(false, a, false, b0, (short)0, acc0, false, false);
        acc1 = __builtin_amdgcn_wmma_f32_16x16x4_f32(false, a, false, b1, (short)0, acc1, false, false);
    }

    const int row0 = mt * 16 + 8 * (lane >> 4);
#pragma unroll
    for (int r = 0; r < 8; ++r) {
        int row = row0 + r;
        if (row < N) {
            H2[(size_t)row * 17 + n16] = acc0[r];       // cols 0..15
            if (n16 == 0) H2[(size_t)row * 17 + 16] = acc1[r];  // col 16 valid
        }
    }
}

// ----------------- attention coefficients layer 2: [N] ---------------------
__global__ void alpha2_kernel(const float* __restrict__ h2,
                              const float* __restrict__ a_src,  // [17]
                              const float* __restrict__ a_dst,  // [17]
                              float* __restrict__ as2, float* __restrict__ ad2,
                              int N) {
    int n = blockIdx.x * blockDim.x + threadIdx.x;
    if (n >= N) return;
    float s = 0.f, d = 0.f;
#pragma unroll
    for (int c = 0; c < 17; ++c) {
        float v = h2[(size_t)n * 17 + c];
        s += v * a_src[c]; d += v * a_dst[c];
    }
    as2[n] = s; ad2[n] = d;
}

__global__ void edge_logits2(const int* __restrict__ ei, int E, int ET,
                             const float* __restrict__ as2,
                             const float* __restrict__ ad2,
                             float* __restrict__ e2, float* __restrict__ m2) {
    int e = blockIdx.x * blockDim.x + threadIdx.x;
    if (e >= ET) return;
    int s, d; edge_sd(ei, E, e, s, d);
    float v = leakyf(as2[s] + ad2[d]);
    e2[e] = v;
    atomicMaxF(&m2[d], v);
}

__global__ void edge_p2(const int* __restrict__ ei, int E, int ET,
                        float* __restrict__ e2, const float* __restrict__ m2,
                        float* __restrict__ d2) {
    int e = blockIdx.x * blockDim.x + threadIdx.x;
    if (e >= ET) return;
    int s, d; edge_sd(ei, E, e, s, d);
    float p = __expf(e2[e] - m2[d]);
    e2[e] = p;
    atomicAdd(&d2[d], p);
}

__global__ void aggregate2(const int* __restrict__ ei, int E, int ET,
                           const float* __restrict__ h2,
                           const float* __restrict__ e2,
                           const float* __restrict__ d2,
                           float* __restrict__ out) {
    int t = blockIdx.x * blockDim.x + threadIdx.x;
    int e = t >> 5, c = t & 31;
    if (e >= ET || c >= 17) return;
    int s, d; edge_sd(ei, E, e, s, d);
    float alpha = e2[e] / (d2[d] + 1e-16f);
    atomicAdd(&out[(size_t)d * 17 + c], alpha * h2[(size_t)s * 17 + c]);
}

// ------------------------- in-place log-softmax over 17 --------------------
__global__ void logsoftmax_kernel(float* __restrict__ out, int N) {
    int n = blockIdx.x * blockDim.x + threadIdx.x;
    if (n >= N) return;
    float* row = out + (size_t)n * 17;
    float mx = row[0];
#pragma unroll
    for (int c = 1; c < 17; ++c) mx = fmaxf(mx, row[c]);
    float s = 0.f;
#pragma unroll
    for (int c = 0; c < 17; ++c) s += __expf(row[c] - mx);
    float l = __logf(s) + mx;
#pragma unroll
    for (int c = 0; c < 17; ++c) row[c] -= l;
}

// --------------------------------- launch ----------------------------------

extern "C" void kernel_launch(void* const* d_in, const int* in_sizes, int n_in,
                              void* d_out, int out_size, void* d_ws, size_t ws_size,
                              hipStream_t stream) {
    const float* x    = (const float*)d_in[0];   // [N,64]
    const int*   ei   = (const int*)d_in[1];     // [2,E]
    const float* W1   = (const float*)d_in[2];   // [64,512]
    const float* as1w = (const float*)d_in[3];   // [8,64]
    const float* ad1w = (const float*)d_in[4];   // [8,64]
    const float* b1   = (const float*)d_in[5];   // [512]
    const float* W2   = (const float*)d_in[6];   // [512,17]
    const float* as2w = (const float*)d_in[7];   // [17]
    const float* ad2w = (const float*)d_in[8];   // [17]
    const float* b2   = (const float*)d_in[9];   // [17]
    float* out = (float*)d_out;

    const int N  = in_sizes[0] / 64;
    const int E  = in_sizes[1] / 2;
    const int ET = E + N;                      // + self loops
    const int Mt = (N + 15) / 16;

    // workspace partition (floats)
    float* p = (float*)d_ws;
    float* h1   = p; p += (size_t)N * 512;
    float* out1 = p; p += (size_t)N * 512;
    float* e1   = p; p += (size_t)ET * 8;
    float* m1   = p; p += (size_t)N * 8;
    float* d1   = p; p += (size_t)N * 8;
    float* asrc = p; p += (size_t)N * 8;
    float* adst = p; p += (size_t)N * 8;
    float* h2   = p; p += (size_t)N * 17;
    float* as2  = p; p += N;
    float* ad2  = p; p += N;
    float* e2   = p; p += ET;
    float* m2   = p; p += N;
    float* d2   = p; p += N;
    float* W2p  = p; p += 512 * 32;

    const int B = 256;
    auto g = [](long long n, int b) { return (int)((n + b - 1) / b); };

    // ---- init ----
    k_fill<<<g((long long)N * 8, B), B, 0, stream>>>(m1, -INFINITY, N * 8);
    k_fill<<<g((long long)N * 8, B), B, 0, stream>>>(d1, 0.f, N * 8);
    k_fill<<<g(N, B), B, 0, stream>>>(m2, -INFINITY, N);
    k_fill<<<g(N, B), B, 0, stream>>>(d2, 0.f, N);
    k_bias_init<<<g((long long)N * 512, B), B, 0, stream>>>(out1, b1, N * 512, 512);
    k_bias_init<<<g((long long)N * 17, B), B, 0, stream>>>(out, b2, N * 17, 17);
    k_pad_w2<<<g(512 * 32, B), B, 0, stream>>>(W2, W2p);

    // ---- layer 1 ----
    {
        int waves = Mt * 8;                    // 8 groups of 4 column tiles
        gemm1_wmma<<<g(waves, 8), B, 0, stream>>>(x, W1, h1, N, Mt);
    }
    alpha1_kernel<<<g((long long)N * 8, B), B, 0, stream>>>(h1, as1w, ad1w, asrc, adst, N);
    edge_logits1<<<g((long long)ET * 8, B), B, 0, stream>>>(ei, E, ET, asrc, adst, e1, m1);
    edge_p1<<<g((long long)ET * 8, B), B, 0, stream>>>(ei, E, ET, e1, m1, d1);
    aggregate1<<<g(ET, 2), B, 0, stream>>>(ei, E, ET, h1, e1, d1, out1);

    // ---- layer 2 ----
    {
        int waves = Mt;                        // one wave covers both col tiles
        gemm2_wmma<<<g(waves, 8), B, 0, stream>>>(out1, W2p, h2, N, Mt);
    }
    alpha2_kernel<<<g(N, B), B, 0, stream>>>(h2, as2w, ad2w, as2, ad2, N);
    edge_logits2<<<g(ET, B), B, 0, stream>>>(ei, E, ET, as2, ad2, e2, m2);
    edge_p2<<<g(ET, B), B, 0, stream>>>(ei, E, ET, e2, m2, d2);
    aggregate2<<<g((long long)ET * 32, B), B, 0, stream>>>(ei, E, ET, h2, e2, d2, out);

    logsoftmax_kernel<<<g(N, B), B, 0, stream>>>(out, N);
}